// RegularizedFMNet_18751827214347
// MI455X (gfx1250) — compile-verified
//
#include <hip/hip_runtime.h>
#include <hip/hip_bf16.h>
#include <math.h>

typedef __attribute__((ext_vector_type(2))) float v2f;
typedef __attribute__((ext_vector_type(8))) float v8f;

#define KDIM 128
#define CDIM 256
#define VDIM 50000
#define LMBDA_ 100.0f
#define VCHUNK 500   // 100 chunks -> 1600 blocks/projection at 32x64 tiles

// ---------------------------------------------------------------------------
// Zero the atomic-accumulation region of workspace (A and B projections).
// ---------------------------------------------------------------------------
__global__ void zero_ws_kernel(float* __restrict__ p, int n) {
  int i = blockIdx.x * blockDim.x + threadIdx.x;
  if (i < n) p[i] = 0.0f;
}

// ---------------------------------------------------------------------------
// out[k, c] += sum_{v in chunk} evecs[v, k] * feat[c, v]
//   evecs: (V, 128) row-major, feat: (256, V) row-major, out: (128, 256)
// One wave computes a 32(k) x 64(c) tile over a V-chunk:
//   2 A-fragments x 4 B-fragments -> 8 v_wmma_f32_16x16x4_f32 per K-step,
//   fed by 4 b32 + 4 b64 loads (wmma:vmem = 1:1), split-V via f32 atomics.
// Fragment mapping follows the CDNA5 32-bit A/B layouts:
//   lanes 0-15 carry K=0,1 in VGPR0/1; lanes 16-31 carry K=2,3.
// ---------------------------------------------------------------------------
__global__ __launch_bounds__(32) void proj_gemm_kernel(
    const float* __restrict__ evecs, const float* __restrict__ feat,
    float* __restrict__ out) {
  const int lane = threadIdx.x & 31;
  const int half = lane >> 4;   // 0: K=0,1   1: K=2,3
  const int l16  = lane & 15;
  const int c0 = blockIdx.x * 64;
  const int k0 = blockIdx.y * 32;
  const int v0 = blockIdx.z * VCHUNK;

  // A fragment element (m=l16, kk=2*half + q): evecs[(v+kk)*128 + (k0+m)]
  const float* ea = evecs + (size_t)(v0 + 2 * half) * KDIM + (k0 + l16);
  // B fragment element (kk=2*half + q, n=l16): feat[(c0+n)*V + (v+kk)]
  const float* fb = feat + (size_t)(c0 + l16) * VDIM + v0 + 2 * half;

  v8f acc[2][4] = {};
  for (int v = 0; v < VCHUNK; v += 4) {
    v2f a[2];
#pragma unroll
    for (int kt = 0; kt < 2; ++kt) {
      a[kt].x = ea[kt * 16];           // (m, kk = 2*half)
      a[kt].y = ea[kt * 16 + KDIM];    // (m, kk = 2*half+1)
    }
    v2f b[4];
#pragma unroll
    for (int ct = 0; ct < 4; ++ct)
      b[ct] = *(const v2f*)(fb + (size_t)ct * 16 * VDIM);  // global_load_b64

#pragma unroll
    for (int kt = 0; kt < 2; ++kt)
#pragma unroll
      for (int ct = 0; ct < 4; ++ct)
        acc[kt][ct] = __builtin_amdgcn_wmma_f32_16x16x4_f32(
            false, a[kt], false, b[ct], (short)0, acc[kt][ct], false, false);

    ea += 4 * KDIM;
    fb += 4;
  }

  // C/D layout: VGPR r holds (M=r, N=lane) for lanes 0-15, (M=r+8) for 16-31.
#pragma unroll
  for (int kt = 0; kt < 2; ++kt) {
#pragma unroll
    for (int ct = 0; ct < 4; ++ct) {
      float* o = out + (size_t)(k0 + 16 * kt + half * 8) * CDIM
                     + (c0 + 16 * ct + l16);
#pragma unroll
      for (int r = 0; r < 8; ++r)
        atomicAdd(o + (size_t)r * CDIM, acc[kt][ct][r]);
    }
  }
}

// ---------------------------------------------------------------------------
// out[i, j] = sum_c X[i, c] * A[j, c]     (X, A: (128,256); out: (128,128))
// One wave per 16x16 tile; inner dim 256 -> 64 WMMA f32 16x16x4 steps.
// Both fragments load as contiguous float2 pairs.
// ---------------------------------------------------------------------------
__global__ __launch_bounds__(32) void gram_kernel(
    const float* __restrict__ X, const float* __restrict__ A,
    float* __restrict__ out) {
  const int lane = threadIdx.x & 31;
  const int half = lane >> 4;
  const int l16  = lane & 15;
  const int j0 = blockIdx.x * 16;
  const int i0 = blockIdx.y * 16;

  const float* xa = X + (size_t)(i0 + l16) * CDIM + 2 * half;
  const float* ab = A + (size_t)(j0 + l16) * CDIM + 2 * half;

  v8f acc = {};
#pragma unroll 4
  for (int c = 0; c < CDIM; c += 4) {
    v2f a = *(const v2f*)(xa + c);
    v2f b = *(const v2f*)(ab + c);
    acc = __builtin_amdgcn_wmma_f32_16x16x4_f32(
        false, a, false, b, (short)0, acc, false, false);
  }

  float* o = out + (size_t)(i0 + half * 8) * KDIM + (j0 + l16);
#pragma unroll
  for (int r = 0; r < 8; ++r)
    o[(size_t)r * KDIM] = acc[r];
}

// ---------------------------------------------------------------------------
// Batched solve: for batch i, (A_A_t + 100*diag(D[i])) x = B_A_t[i,:].
// One workgroup (128 threads) per batch; augmented 128x129 system in LDS
// (stride 129 -> conflict-free banks). Gaussian elimination + back-subst.
// System is SPD (Gram + nonneg diagonal) -> no pivoting needed.
// ---------------------------------------------------------------------------
__global__ __launch_bounds__(128) void solve_kernel(
    const float* __restrict__ AAt, const float* __restrict__ BAt,
    const float* __restrict__ evx, const float* __restrict__ evy,
    float* __restrict__ out) {
  __shared__ float M[KDIM * (KDIM + 1)];
  __shared__ float red[KDIM];
  __shared__ float sm;
  const int t = threadIdx.x;
  const int i = blockIdx.x;  // batch index (row of D -> evals_y)

  // s = max(max(evals_x), max(evals_y))
  red[t] = fmaxf(evx[t], evy[t]);
  __syncthreads();
  for (int w = 64; w > 0; w >>= 1) {
    if (t < w) red[t] = fmaxf(red[t], red[t + w]);
    __syncthreads();
  }
  if (t == 0) sm = red[0];
  __syncthreads();
  const float s = sm;

  // Load A_A_t into LDS (coalesced over flat index).
  for (int idx = t; idx < KDIM * KDIM; idx += KDIM) {
    int r = idx >> 7, c = idx & 127;
    M[r * 129 + c] = AAt[idx];
  }
  __syncthreads();

  // Resolvent-mask diagonal + RHS; thread t owns row t.
  {
    float g2 = sqrtf(evy[i] / s);      // gamma = 0.5
    float g1 = sqrtf(evx[t] / s);
    float d2 = g2 * g2 + 1.0f, d1 = g1 * g1 + 1.0f;
    float mre = g2 / d2 - g1 / d1;
    float mim = 1.0f / d2 - 1.0f / d1;
    float D = mre * mre + mim * mim;
    M[t * 129 + t] += LMBDA_ * D;
    M[t * 129 + 128] = BAt[i * KDIM + t];
  }
  __syncthreads();

  // Forward elimination (no pivoting).
  for (int p = 0; p < KDIM - 1; ++p) {
    float inv = 1.0f / M[p * 129 + p];
    if (t > p) {
      float f = M[t * 129 + p] * inv;
      for (int j = p + 1; j <= KDIM; ++j)
        M[t * 129 + j] -= f * M[p * 129 + j];
    }
    __syncthreads();
  }

  // Back substitution.
  for (int p = KDIM - 1; p >= 0; --p) {
    if (t == p) M[p * 129 + 128] /= M[p * 129 + p];
    __syncthreads();
    if (t < p) M[t * 129 + 128] -= M[t * 129 + p] * M[p * 129 + 128];
    __syncthreads();
  }

  out[(size_t)i * KDIM + t] = M[t * 129 + 128];
}

// ---------------------------------------------------------------------------
extern "C" void kernel_launch(void* const* d_in, const int* in_sizes, int n_in,
                              void* d_out, int out_size, void* d_ws,
                              size_t ws_size, hipStream_t stream) {
  const float* feat_x  = (const float*)d_in[0];  // (256, 50000)
  const float* feat_y  = (const float*)d_in[1];  // (256, 50000)
  const float* evals_x = (const float*)d_in[2];  // (128,)
  const float* evecs_x = (const float*)d_in[3];  // (50000, 128)
  const float* evals_y = (const float*)d_in[4];  // (128,)
  const float* evecs_y = (const float*)d_in[5];  // (50000, 128)
  float* out = (float*)d_out;                    // (1, 128, 128)

  float* A   = (float*)d_ws;           // (128, 256)
  float* B   = A + KDIM * CDIM;        // (128, 256)
  float* AAt = B + KDIM * CDIM;        // (128, 128)
  float* BAt = AAt + KDIM * KDIM;      // (128, 128)

  // Zero atomic accumulation targets (required every call).
  const int nz = 2 * KDIM * CDIM;
  zero_ws_kernel<<<(nz + 255) / 256, 256, 0, stream>>>(A, nz);

  // Big memory-bound projections, split-V with f32 atomics, 32x64 wave tiles.
  dim3 gproj(CDIM / 64, KDIM / 32, VDIM / VCHUNK);
  proj_gemm_kernel<<<gproj, 32, 0, stream>>>(evecs_x, feat_x, A);
  proj_gemm_kernel<<<gproj, 32, 0, stream>>>(evecs_y, feat_y, B);

  // Gram matrices.
  dim3 ggram(KDIM / 16, KDIM / 16);
  gram_kernel<<<ggram, 32, 0, stream>>>(A, A, AAt);
  gram_kernel<<<ggram, 32, 0, stream>>>(B, A, BAt);

  // Batched regularized solves.
  solve_kernel<<<KDIM, KDIM, 0, stream>>>(AAt, BAt, evals_x, evals_y, out);
}